// RewireGearnet_61297773248646
// MI455X (gfx1250) — compile-verified
//
#include <hip/hip_runtime.h>
#include <hip/hip_bf16.h>

// Problem constants (from reference)
#define NN      100000
#define RR      7
#define DD      64
#define EE      1600000
#define EDIM    16
#define NR      (NN * RR)          // 700000 segments
#define UPD_F   ((size_t)NR * DD)  // 44,800,000 floats = 179.2 MB (lives in d_ws)

typedef float v2f __attribute__((ext_vector_type(2)));
typedef float v8f __attribute__((ext_vector_type(8)));

// ---------------------------------------------------------------------------
// Kernel 1: zero the segment-sum buffer (float4 stores, exact coverage)
// ---------------------------------------------------------------------------
__global__ void zero_update_kernel(float4* __restrict__ upd4) {
    size_t i = (size_t)blockIdx.x * blockDim.x + threadIdx.x;  // UPD_F/4 threads
    upd4[i] = make_float4(0.f, 0.f, 0.f, 0.f);
}

// ---------------------------------------------------------------------------
// Kernel 2: edge scatter.  64 threads per edge, thread d owns output dim d.
//   val = ew * ( x[src][d] + (W_edge[d,:] . ef) + b_edge[d] )
//   atomicAdd(update[(dst*R + rel)*64 + d], val)   -- coalesced 64-lane runs
// ---------------------------------------------------------------------------
__global__ void edge_scatter_kernel(const float* __restrict__ x,
                                    const int*   __restrict__ edge_list,
                                    const float* __restrict__ edge_weight,
                                    const float* __restrict__ edge_feature,
                                    const float* __restrict__ W_edge,
                                    const float* __restrict__ b_edge,
                                    float*       __restrict__ update) {
    const int e = blockIdx.x * 4 + (threadIdx.x >> 6);   // 4 edges / 256-thr block
    const int d = threadIdx.x & 63;

    const int src = edge_list[e * 3 + 0];
    const int dst = edge_list[e * 3 + 1];
    const int rel = edge_list[e * 3 + 2];
    const float ew = edge_weight[e];

    // edge MLP row: 16-wide dot (W_edge is 64x16 row-major, 4 KB -> L0 resident)
    const float* wrow = W_edge + d * EDIM;
    const float* ef   = edge_feature + (size_t)e * EDIM;
    float emsg = b_edge[d];
#pragma unroll
    for (int k = 0; k < EDIM; ++k) emsg = fmaf(wrow[k], ef[k], emsg);

    const float xv  = x[(size_t)src * DD + d];
    const float val = ew * (xv + emsg);

    atomicAdd(&update[((size_t)dst * RR + rel) * DD + d], val);
}

// ---------------------------------------------------------------------------
// Kernel 3: node GEMM with V_WMMA_F32_16X16X4_F32.
//   out(16 x 64 per block) = relu( [update(16x448) | x(16x64)] * Wcat^T + bias )
//   Wcat(d, k) = k < 448 ? W_lin[d,k] : W_self[d, k-448]
// Block = 128 threads (4 waves); wave w computes output columns [16w, 16w+16).
// A tile staged in LDS, row stride 516 floats (516 % 64 == 4 -> bank-clean
// ds_load_b64 in the A-fragment pattern: lane l hits banks 4l..4l+3).
// ---------------------------------------------------------------------------
#define ASTR 516

__global__ void __launch_bounds__(128)
node_gemm_wmma_kernel(const float* __restrict__ update,
                      const float* __restrict__ x,
                      const float* __restrict__ W_lin,
                      const float* __restrict__ b_lin,
                      const float* __restrict__ W_self,
                      const float* __restrict__ b_self,
                      float*       __restrict__ out) {
    __shared__ float sA[16 * ASTR];          // 33 KB

    const int tid       = threadIdx.x;
    const int node_base = blockIdx.x * 16;

    // ---- cooperative fill: 16 rows x 512 cols, float4 granularity ----------
#pragma unroll 4
    for (int v = tid; v < 16 * 128; v += 128) {
        const int row = v >> 7;
        const int col = (v & 127) * 4;
        float4 val;
        if (col < 448) {
            val = *reinterpret_cast<const float4*>(
                      update + (size_t)(node_base + row) * 448 + col);
        } else {
            val = *reinterpret_cast<const float4*>(
                      x + (size_t)(node_base + row) * DD + (col - 448));
        }
        *reinterpret_cast<float4*>(&sA[row * ASTR + col]) = val;
    }
    __syncthreads();

    // ---- per-wave WMMA accumulation ----------------------------------------
    const int lane = tid & 31;
    const int wv   = tid >> 5;
    const int n0   = wv * 16;                 // this wave's output-column base
    const int arow = lane & 15;               // A-matrix M row for this lane
    const int koff = (lane < 16) ? 0 : 2;     // K sub-offset per half-wave

    const float* sArow = &sA[arow * ASTR + koff];
    const float* wlrow = W_lin  + (size_t)(n0 + arow) * 448 + koff;  // B frag rows
    const float* wsrow = W_self + (size_t)(n0 + arow) * DD  + koff;

    v8f c = {0.f, 0.f, 0.f, 0.f, 0.f, 0.f, 0.f, 0.f};

#pragma unroll 8
    for (int k = 0; k < 448; k += 4) {        // 112 WMMAs over W_lin
        v2f a = *reinterpret_cast<const v2f*>(sArow + k);
        v2f b = *reinterpret_cast<const v2f*>(wlrow + k);
        c = __builtin_amdgcn_wmma_f32_16x16x4_f32(
                false, a, false, b, (short)0, c, false, false);
    }
#pragma unroll
    for (int k = 0; k < 64; k += 4) {         // 16 WMMAs over W_self
        v2f a = *reinterpret_cast<const v2f*>(sArow + 448 + k);
        v2f b = *reinterpret_cast<const v2f*>(wsrow + k);
        c = __builtin_amdgcn_wmma_f32_16x16x4_f32(
                false, a, false, b, (short)0, c, false, false);
    }

    // ---- epilogue: bias + ReLU + coalesced row stores ----------------------
    const int dcol  = n0 + (lane & 15);       // output feature index
    const float bias = b_lin[dcol] + b_self[dcol];
    const int rbase  = node_base + ((lane < 16) ? 0 : 8);
#pragma unroll
    for (int v = 0; v < 8; ++v) {
        float val = c[v] + bias;
        val = fmaxf(val, 0.f);
        out[(size_t)(rbase + v) * DD + dcol] = val;
    }
}

// ---------------------------------------------------------------------------
// launch
// ---------------------------------------------------------------------------
extern "C" void kernel_launch(void* const* d_in, const int* in_sizes, int n_in,
                              void* d_out, int out_size, void* d_ws, size_t ws_size,
                              hipStream_t stream) {
    const float* x            = (const float*)d_in[0];
    const int*   edge_list    = (const int*)  d_in[1];
    const float* edge_weight  = (const float*)d_in[2];
    const float* edge_feature = (const float*)d_in[3];
    const float* W_lin        = (const float*)d_in[4];
    const float* b_lin        = (const float*)d_in[5];
    const float* W_self       = (const float*)d_in[6];
    const float* b_self       = (const float*)d_in[7];
    const float* W_edge       = (const float*)d_in[8];
    const float* b_edge       = (const float*)d_in[9];
    float*       out          = (float*)d_out;
    float*       update       = (float*)d_ws;   // needs 179.2 MB of workspace

    // 1) zero segment-sum buffer: 11.2M float4 / 256 = 43750 blocks (exact)
    zero_update_kernel<<<(unsigned)(UPD_F / 4 / 256), 256, 0, stream>>>(
        (float4*)update);

    // 2) edge scatter: 4 edges per 256-thread block
    edge_scatter_kernel<<<EE / 4, 256, 0, stream>>>(
        x, edge_list, edge_weight, edge_feature, W_edge, b_edge, update);

    // 3) fused node GEMM + bias + ReLU: 16 nodes per block
    node_gemm_wmma_kernel<<<NN / 16, 128, 0, stream>>>(
        update, x, W_lin, b_lin, W_self, b_self, out);
}